// OriginalImplementationPNP_43404939493719
// MI455X (gfx1250) — compile-verified
//
#include <hip/hip_runtime.h>
#include <hip/hip_bf16.h>

// ---------------------------------------------------------------------------
// PNP-D_q loss, N=384, D=512.
// Stage 1: sim = batch @ batch^T with V_WMMA_F32_16X16X4_F32 (exact f32 WMMA).
// Stage 2: per-row masked sigmoid-rank + 1/(1+S)^4, LDS-staged, deterministic
//          tree reduction.
// Stage 3: fixed-order scalar reduction -> 1 - loss.
// ---------------------------------------------------------------------------

#define NN 384
#define DD 512
#define TILES_N 24   // 384 / 16
#define ANNEAL_INV 100.0f

typedef float v2f __attribute__((ext_vector_type(2)));
typedef float v8f __attribute__((ext_vector_type(8)));

// One wave (32 lanes) computes one 16x16 tile of sim = A * A^T.
// A fragment (16x4 f32, 2 VGPRs): lane = m + 16*(k>>1); vgpr = k&1.
// B fragment (4x16 f32, 2 VGPRs): lane = n + 16*(k>>1); vgpr = k&1.
//   For B = A^T: B[k][n] = A[n][k] -> same load shape as A on row base n.
// C/D (16x16 f32, 8 VGPRs): vgpr r -> M = r + 8*(lane>>4), N = lane&15.
__global__ void __launch_bounds__(128)
sim_wmma_kernel(const float* __restrict__ A, float* __restrict__ sim) {
    const int wave = threadIdx.x >> 5;
    const int lane = threadIdx.x & 31;
    const int tile = blockIdx.x * 4 + wave;       // 0..575, grid covers exactly
    const int tm = tile / TILES_N;
    const int tn = tile % TILES_N;

    const int mn_lane = lane & 15;                // row index within fragment
    const int khalf   = (lane >> 4) << 1;         // 0 or 2

    const float* __restrict__ arow = A + (tm * 16 + mn_lane) * DD + khalf;
    const float* __restrict__ brow = A + (tn * 16 + mn_lane) * DD + khalf;

    v8f c = {0.f, 0.f, 0.f, 0.f, 0.f, 0.f, 0.f, 0.f};

    for (int k = 0; k < DD; k += 4) {
        v2f a, b;
        a.x = arow[k];
        a.y = arow[k + 1];
        b.x = brow[k];
        b.y = brow[k + 1];
        // D = A(16x4) * B(4x16) + C
        c = __builtin_amdgcn_wmma_f32_16x16x4_f32(
                /*neg_a=*/false, a, /*neg_b=*/false, b,
                /*c_mod=*/(short)0, c, /*reuse_a=*/false, /*reuse_b=*/false);
    }

    const int row0 = tm * 16 + ((lane >> 4) << 3);   // M base for this lane half
    const int col  = tn * 16 + mn_lane;
#pragma unroll
    for (int r = 0; r < 8; ++r) {
        sim[(row0 + r) * NN + col] = c[r];
    }
}

// One block per row i. S_ij = sum_k [lab_k != lab_i] * sigmoid((sim_ik-sim_ij)*100)
// row_loss[i] = sum_{j pos} (1+S_ij)^-4 / max(N_pos_i, 1)
__global__ void __launch_bounds__(256)
row_loss_kernel(const float* __restrict__ sim, const int* __restrict__ labels,
                float* __restrict__ row_loss) {
    __shared__ float srow[NN];
    __shared__ int   slab[NN];
    __shared__ float redf[256];
    __shared__ int   redi[256];

    const int i = blockIdx.x;
    const int t = threadIdx.x;

    for (int k = t; k < NN; k += 256) {
        srow[k] = sim[i * NN + k];
        slab[k] = labels[k];
    }
    __syncthreads();

    const int li = slab[i];

    float contrib = 0.0f;
    int   poscnt  = 0;

    for (int j = t; j < NN; j += 256) {
        if (j != i && slab[j] == li) {
            ++poscnt;
            const float sij = srow[j];
            float S = 0.0f;
            for (int k = 0; k < NN; ++k) {
                if (slab[k] != li) {
                    float e = (sij - srow[k]) * ANNEAL_INV;
                    e = fminf(fmaxf(e, -50.0f), 50.0f);
                    S += 1.0f / (1.0f + __expf(e));
                }
            }
            const float d  = 1.0f + S;
            const float d2 = d * d;
            contrib += 1.0f / (d2 * d2);       // (1+S)^-ALPHA, ALPHA=4
        }
    }

    redf[t] = contrib;
    redi[t] = poscnt;
    __syncthreads();
#pragma unroll
    for (int s = 128; s > 0; s >>= 1) {
        if (t < s) {
            redf[t] += redf[t + s];
            redi[t] += redi[t + s];
        }
        __syncthreads();
    }
    if (t == 0) {
        row_loss[i] = redf[0] / fmaxf((float)redi[0], 1.0f);
    }
}

__global__ void finalize_kernel(const float* __restrict__ row_loss,
                                float* __restrict__ out) {
    if (threadIdx.x == 0) {
        float s = 0.0f;
        for (int r = 0; r < NN; ++r) s += row_loss[r];   // fixed order: deterministic
        out[0] = 1.0f - s * (1.0f / (float)NN);
    }
}

extern "C" void kernel_launch(void* const* d_in, const int* in_sizes, int n_in,
                              void* d_out, int out_size, void* d_ws, size_t ws_size,
                              hipStream_t stream) {
    (void)in_sizes; (void)n_in; (void)out_size; (void)ws_size;

    const float* batch  = (const float*)d_in[0];
    const int*   labels = (const int*)d_in[1];

    float* sim      = (float*)d_ws;          // 384*384 f32 = 576 KiB
    float* row_loss = sim + NN * NN;         // 384 f32

    // 576 tiles (24x24), 4 waves per 128-thread block -> 144 blocks
    sim_wmma_kernel<<<144, 128, 0, stream>>>(batch, sim);
    row_loss_kernel<<<NN, 256, 0, stream>>>(sim, labels, row_loss);
    finalize_kernel<<<1, 32, 0, stream>>>(row_loss, (float*)d_out);
}